// CrossViewFusionModule_50062138802507
// MI455X (gfx1250) — compile-verified
//
#include <hip/hip_runtime.h>

// CrossViewFusion for MI455X (gfx1250, wave32).
// B=16, D=256, W=H=128. Memory-bound: ~768 MB HBM traffic floor -> ~33us @
// 23.3 TB/s; reverse-order second pass recovers up to ~192 MB from L2.
// Score GEMV runs on the matrix pipe via V_WMMA_F32_16X16X4_F32 (fp32-exact),
// sum-of-squares runs concurrently on VALU; context pass streams with NT hints.

#define BB 16
#define DD 256
#define WH 16384  // 128*128

typedef __attribute__((ext_vector_type(2))) float v2f;
typedef __attribute__((ext_vector_type(8))) float v8f;

// ---------------------------------------------------------------- kernel 1
// q_hat = q / max(||q||, 1e-12). One wave per batch row (16 waves, 1 block).
__global__ void cvf_qnorm(const float* __restrict__ q, float* __restrict__ qh) {
  const int lane = threadIdx.x & 31;
  const int b    = threadIdx.x >> 5;  // 0..15
  float v[8];
  float ssq = 0.f;
#pragma unroll
  for (int i = 0; i < 8; ++i) {
    v[i] = q[b * DD + i * 32 + lane];
    ssq = fmaf(v[i], v[i], ssq);
  }
#pragma unroll
  for (int off = 16; off >= 1; off >>= 1) ssq += __shfl_xor(ssq, off, 32);
  const float inv = 1.f / fmaxf(sqrtf(ssq), 1e-12f);
#pragma unroll
  for (int i = 0; i < 8; ++i) qh[b * DD + i * 32 + lane] = v[i] * inv;
}

// ---------------------------------------------------------------- kernel 2
// Per wave: 16 wh columns of one batch. Accumulate over D in K=4 WMMA steps.
//   A (16x4 f32): row 0 = q_hat[d..d+3], rows 1..15 = 0.
//     Layout: VGPR0 = K0 (lanes 0-15) / K2 (lanes 16-31); VGPR1 = K1 / K3.
//     => lane 0 carries (q[d],q[d+1]), lane 16 carries (q[d+2],q[d+3]).
//     Every lane loads the b64 pair q[d+kBase..+1] (2 cachelines per wave,
//     L0 hits) and masks with the branch-free (col==0) cndmask.
//   B (4x16 f32): lanes = column n, lane half selects K pair.
//   D (16x16 f32): VGPR0, lanes 0-15 = row M=0 = the 16 dot products.
// ssq: lane n accumulates K%4 in {0,1}, lane n+16 accumulates K%4 in {2,3};
// shfl_xor(16) completes the 256-term sum per column.
__global__ void cvf_score(const float* __restrict__ value,
                          const float* __restrict__ qh,
                          float* __restrict__ score,
                          float* __restrict__ invn) {
  const int lane  = threadIdx.x & 31;
  const int wave  = threadIdx.x >> 5;       // 8 waves / block
  const int b     = blockIdx.x >> 7;        // 128 blocks per batch
  const int chunk = blockIdx.x & 127;
  const int wh0   = chunk * 128 + wave * 16;
  const int col   = lane & 15;
  const int kBase = (lane >> 4) << 1;       // 0 (lanes 0-15) or 2 (lanes 16-31)
  const bool isA  = (col == 0);             // lanes 0 and 16 carry A row 0

  const float* __restrict__ vbase = value + (size_t)b * DD * WH + wh0 + col;
  const float* __restrict__ qb    = qh + b * DD + kBase;  // 8-byte aligned

  v8f   acc = {};
  float ssq = 0.f;
#pragma unroll 4
  for (int d = 0; d < DD; d += 4) {
    const float b0 = vbase[(size_t)(d + kBase) * WH];
    const float b1 = vbase[(size_t)(d + kBase + 1) * WH];
    v2f bm; bm.x = b0; bm.y = b1;

    const v2f qv = *(const v2f*)(qb + d);   // one global_load_b64, cached
    v2f am;
    am.x = isA ? qv.x : 0.f;                // v_cndmask, no branches
    am.y = isA ? qv.y : 0.f;

    // 8 args: (neg_a, A, neg_b, B, c_mod, C, reuse_a, reuse_b)
    acc = __builtin_amdgcn_wmma_f32_16x16x4_f32(
        /*neg_a=*/false, am, /*neg_b=*/false, bm,
        /*c_mod=*/(short)0, acc, /*reuse_a=*/false, /*reuse_b=*/false);

    ssq = fmaf(b0, b0, ssq);
    ssq = fmaf(b1, b1, ssq);
  }

  const float ssq_tot = ssq + __shfl_xor(ssq, 16, 32);
  if (lane < 16) {
    const float denom = fmaxf(sqrtf(ssq_tot), 1e-12f);
    const size_t idx  = (size_t)b * WH + wh0 + col;
    score[idx] = acc[0] / denom;   // D row M=0, N=lane
    invn[idx]  = 1.f / denom;
  }
}

// ---------------------------------------------------------------- kernel 3
// Per-batch min/max over 16384 scores. One block per batch.
__global__ void cvf_minmax(const float* __restrict__ score, float* __restrict__ mm) {
  __shared__ float smin[256];
  __shared__ float smax[256];
  const int b = blockIdx.x, t = threadIdx.x;
  float mn = __builtin_inff(), mx = -__builtin_inff();
  for (int i = t; i < WH; i += 256) {
    const float s = score[(size_t)b * WH + i];
    mn = fminf(mn, s);
    mx = fmaxf(mx, s);
  }
  smin[t] = mn; smax[t] = mx;
  __syncthreads();
  for (int off = 128; off >= 1; off >>= 1) {
    if (t < off) {
      smin[t] = fminf(smin[t], smin[t + off]);
      smax[t] = fmaxf(smax[t], smax[t + off]);
    }
    __syncthreads();
  }
  if (t == 0) { mm[2 * b] = smin[0]; mm[2 * b + 1] = smax[0]; }
}

// ---------------------------------------------------------------- kernel 4
// attn = (s-min)/(max-min+1e-8); context = attn * inv_norm * value.
// Blocks run in REVERSE order: cvf_score streamed value front-to-back, so L2
// (192 MB of the 256 MB stream) holds the tail -- reading tail-first turns
// most of the re-read into L2 hits instead of HBM traffic.
// NT loads/stores (stream > L2) + global_prefetch_b8 ahead of the load.
__global__ void cvf_context(const float* __restrict__ value,
                            const float* __restrict__ score,
                            const float* __restrict__ invn,
                            const float* __restrict__ mm,
                            float* __restrict__ ctx,
                            float* __restrict__ attn_out) {
  const int rb  = gridDim.x - 1 - blockIdx.x;      // reversed block order
  const int tid = rb * 256 + threadIdx.x;          // 0 .. B*WH-1
  const int b   = tid >> 14;                       // WH = 2^14
  const int wh  = tid & (WH - 1);

  const float mn = mm[2 * b], mx = mm[2 * b + 1];
  const float a  = (score[tid] - mn) / (mx - mn + 1e-8f);
  attn_out[tid]  = a;
  const float scale = a * invn[tid];

  const float* __restrict__ src = value + (size_t)b * DD * WH + wh;
  float* __restrict__ dst       = ctx + (size_t)b * DD * WH + wh;
#pragma unroll 4
  for (int d = 0; d < DD - 8; ++d) {
    __builtin_prefetch(src + (size_t)(d + 8) * WH, 0, 1);
    const float v = __builtin_nontemporal_load(src + (size_t)d * WH);
    __builtin_nontemporal_store(v * scale, dst + (size_t)d * WH);
  }
#pragma unroll
  for (int d = DD - 8; d < DD; ++d) {
    const float v = __builtin_nontemporal_load(src + (size_t)d * WH);
    __builtin_nontemporal_store(v * scale, dst + (size_t)d * WH);
  }
}

// ---------------------------------------------------------------- launch
extern "C" void kernel_launch(void* const* d_in, const int* in_sizes, int n_in,
                              void* d_out, int out_size, void* d_ws, size_t ws_size,
                              hipStream_t stream) {
  const float* q     = (const float*)d_in[0];  // [16, 256]
  const float* value = (const float*)d_in[1];  // [16, 256, 128, 128]

  float* ctx      = (float*)d_out;                         // 16*256*128*128
  float* attn_out = (float*)d_out + (size_t)BB * DD * WH;  // + 16*128*128

  // Workspace: qh | score | invn | minmax  (~2.07 MB)
  float* qh    = (float*)d_ws;
  float* score = qh + BB * DD;
  float* invn  = score + (size_t)BB * WH;
  float* mm    = invn + (size_t)BB * WH;

  cvf_qnorm<<<1, 512, 0, stream>>>(q, qh);
  cvf_score<<<BB * 128, 256, 0, stream>>>(value, qh, score, invn);
  cvf_minmax<<<BB, 256, 0, stream>>>(score, mm);
  cvf_context<<<(BB * WH) / 256, 256, 0, stream>>>(value, score, invn, mm,
                                                   ctx, attn_out);
}